// SSDLoss_22454089023657
// MI455X (gfx1250) — compile-verified
//
#include <hip/hip_runtime.h>

// ---------------- problem constants ----------------
#define B_      32
#define NA_     32768
#define NC_     81
#define NG_     24
#define NPR_    3
#define POS_IOU_ 0.5f
#define NEG_IOU_ 0.4f
#define VXY_    0.1f
#define VWH_    0.2f

// ---------------- CE streaming kernel tiling ----------------
#define TILE_A   128                       // anchors per tile (1 per thread)
#define THR_C    128                       // threads in ce kernel
#define T_TILES  4                         // tiles per block
#define NBLK_    (NA_/(TILE_A*T_TILES))    // 64 blocks per batch
#define TILE_F   (TILE_A*NC_)              // 10368 floats per tile
#define TILE_V16 (TILE_F/4)                // 2592 16-byte chunks
#define CP_ITERS ((TILE_V16+THR_C-1)/THR_C)// 21 async loads per thread (uniform!)

// ---------------- CDNA5 async global->LDS path (guarded) ----------------
#if defined(__HIP_DEVICE_COMPILE__) && defined(__gfx1250__) && __has_builtin(__builtin_amdgcn_global_load_async_to_lds_b128)
  #define USE_ASYNC_LDS 1
#endif
#if defined(__HIP_DEVICE_COMPILE__) && defined(__gfx1250__) && __has_builtin(__builtin_amdgcn_s_wait_asynccnt)
  #define HAS_WAIT_ASYNC 1
#endif

typedef int v4i_t __attribute__((vector_size(4 * sizeof(int))));

__device__ __forceinline__ void tile_copy(float* __restrict__ dst,
                                          const float* __restrict__ src, int tid) {
  // copy TILE_V16 16B chunks with THR_C threads; wrap duplicate indices so each
  // wave issues exactly CP_ITERS async-load instructions (uniform ASYNCcnt).
#if defined(USE_ASYNC_LDS)
  #pragma unroll
  for (int k = 0; k < CP_ITERS; ++k) {
    int idx = k * THR_C + tid;
    if (idx >= TILE_V16) idx -= TILE_V16;   // duplicate copy of same data: benign
    float* s = const_cast<float*>(src) + idx * 4;
    float* d = dst + idx * 4;
    __builtin_amdgcn_global_load_async_to_lds_b128((v4i_t*)s, (v4i_t*)d, 0, 0);
  }
#else
  for (int k = 0; k < CP_ITERS; ++k) {
    int idx = k * THR_C + tid;
    if (idx >= TILE_V16) idx -= TILE_V16;
    float4 v = ((const float4*)src)[idx];
    ((float4*)dst)[idx] = v;
  }
#endif
}

__device__ __forceinline__ void wait_async_tile(bool next_in_flight) {
#if defined(USE_ASYNC_LDS) && defined(HAS_WAIT_ASYNC)
  if (next_in_flight) __builtin_amdgcn_s_wait_asynccnt(CP_ITERS); // current tile done
  else                __builtin_amdgcn_s_wait_asynccnt(0);
#else
  (void)next_in_flight;
#endif
}

__device__ __forceinline__ float sl1f(float d) {
  float ad = fabsf(d);
  return ad < 1.f ? 0.5f * d * d : ad - 0.5f;
}

// ---------------- kernel 0: zero best-anchor keys ----------------
__global__ void init_kernel(unsigned long long* best) {
  for (int i = threadIdx.x; i < B_ * NG_; i += blockDim.x) best[i] = 0ull;
}

// ---------------- kernel 1: anchor-GT matching ----------------
__global__ __launch_bounds__(256)
void match_kernel(const float* __restrict__ anchors, const float* __restrict__ gt_boxes,
                  unsigned long long* __restrict__ best, unsigned char* __restrict__ meta) {
  __shared__ float g4[NG_][4];                 // GT boxes in xyxy
  __shared__ unsigned long long red[256];
  const int b = blockIdx.y, tid = threadIdx.x;
  const int a = blockIdx.x * 256 + tid;
  if (tid < NG_) {
    float cx = gt_boxes[(b*NG_+tid)*4+0], cy = gt_boxes[(b*NG_+tid)*4+1];
    float w  = gt_boxes[(b*NG_+tid)*4+2], h  = gt_boxes[(b*NG_+tid)*4+3];
    g4[tid][0] = cx - 0.5f*w; g4[tid][1] = cy - 0.5f*h;
    g4[tid][2] = cx + 0.5f*w; g4[tid][3] = cy + 0.5f*h;
  }
  __syncthreads();
  float acx = anchors[a*4+0], acy = anchors[a*4+1];
  float aw  = anchors[a*4+2], ah  = anchors[a*4+3];
  float ax0 = acx - 0.5f*aw, ay0 = acy - 0.5f*ah;
  float ax1 = acx + 0.5f*aw, ay1 = acy + 0.5f*ah;
  float area_a = (ax1-ax0)*(ay1-ay0);
  float iou[NG_];
  float best_iou = -1.f; int best_g = 0;
  #pragma unroll
  for (int g = 0; g < NG_; ++g) {
    float ltx = fmaxf(ax0, g4[g][0]), lty = fmaxf(ay0, g4[g][1]);
    float rbx = fminf(ax1, g4[g][2]), rby = fminf(ay1, g4[g][3]);
    float w = fmaxf(rbx-ltx, 0.f), h = fmaxf(rby-lty, 0.f);
    float inter = w*h;
    float area_g = (g4[g][2]-g4[g][0])*(g4[g][3]-g4[g][1]);
    float u = fmaxf(area_a + area_g - inter, 1e-12f);
    iou[g] = inter / u;
    if (iou[g] > best_iou) { best_iou = iou[g]; best_g = g; } // first-max tie rule
  }
  int state = (best_iou >= POS_IOU_) ? 2 : ((best_iou > NEG_IOU_) ? 1 : 0);
  meta[(long)b*NA_ + a] = (unsigned char)((best_g << 2) | state);
  // per-GT argmax over anchors: key = iou_bits<<32 | (~a) -> lowest anchor wins ties
  for (int g = 0; g < NG_; ++g) {
    unsigned long long key =
        (((unsigned long long)__float_as_uint(iou[g])) << 32) |
        (unsigned long long)(0xFFFFFFFFu - (unsigned)a);
    red[tid] = key; __syncthreads();
    for (int s = 128; s > 0; s >>= 1) {
      if (tid < s) { unsigned long long o = red[tid+s]; if (o > red[tid]) red[tid] = o; }
      __syncthreads();
    }
    if (tid == 0) atomicMax(&best[b*NG_ + g], red[0]);
    __syncthreads();
  }
}

// ---------------- kernel 2: force best anchors positive ----------------
__global__ void override_kernel(const unsigned long long* __restrict__ best,
                                unsigned int* __restrict__ meta_w) {
  int i = blockIdx.x * blockDim.x + threadIdx.x;
  if (i >= B_ * NG_) return;
  unsigned a = 0xFFFFFFFFu - (unsigned)(best[i] & 0xFFFFFFFFull);
  int b = i / NG_;
  unsigned idx = (unsigned)b * NA_ + a;
  atomicOr(&meta_w[idx >> 2], 2u << (8 * (idx & 3)));   // set pos bit in meta byte
}

// ---------------- kernel 3: streaming CE + smooth-L1 (async LDS pipeline) ----------------
__global__ __launch_bounds__(THR_C)
void ce_kernel(const float* __restrict__ cls_pred, const float* __restrict__ box_pred,
               const float* __restrict__ anchors, const float* __restrict__ gt_boxes,
               const int* __restrict__ gt_labels, const unsigned char* __restrict__ meta,
               float* __restrict__ ce_sel,
               float* __restrict__ p_posce, float* __restrict__ p_reg,
               int* __restrict__ p_npos, int* __restrict__ p_nneg) {
  extern __shared__ float lds[];               // 2 * TILE_F floats (double buffer)
  __shared__ float redf[THR_C];
  __shared__ int   redi[THR_C];
  const int b = blockIdx.y, blk = blockIdx.x, tid = threadIdx.x;
  const long base_a = (long)blk * (TILE_A * T_TILES);
  const float* gbase = cls_pred + ((long)b * NA_ + base_a) * NC_;

  tile_copy(lds, gbase, tid);                  // prime the pipeline
  float acc_pce = 0.f, acc_reg = 0.f; int acc_np = 0, acc_nn = 0;

  for (int t = 0; t < T_TILES; ++t) {
    float* buf = lds + (t & 1) * TILE_F;
    bool have_next = (t + 1 < T_TILES);
    if (have_next) tile_copy(lds + ((t + 1) & 1) * TILE_F,
                             gbase + (long)(t + 1) * TILE_F, tid);
    wait_async_tile(have_next);
    __syncthreads();                           // tile t visible to all waves

    const long a = base_a + (long)t * TILE_A + tid;
    const float* row = buf + tid * NC_;        // stride 81 words -> conflict-free
    unsigned m = meta[(long)b * NA_ + a];
    int state = m & 3, gtidx = (int)(m >> 2);
    bool pos = (state & 2) != 0, neg = (state == 0);
    int tgt = pos ? gt_labels[b * NG_ + gtidx] : 0;

    float mx = row[0];
    #pragma unroll 9
    for (int c = 1; c < NC_; ++c) mx = fmaxf(mx, row[c]);
    float s = 0.f;
    #pragma unroll 9
    for (int c = 0; c < NC_; ++c) s += __expf(row[c] - mx);
    float ce = (mx + __logf(s)) - row[tgt];

    ce_sel[(long)b * NA_ + a] = neg ? fmaxf(ce, 0.f) : 0.f;  // selection key, >= 0
    if (pos) {
      acc_pce += ce; acc_np += 1;
      float4 an = ((const float4*)anchors)[a];
      float4 gb = ((const float4*)gt_boxes)[b * NG_ + gtidx];
      float4 bp = ((const float4*)box_pred)[(long)b * NA_ + a];
      float tx = (gb.x - an.x) / an.z / VXY_;
      float ty = (gb.y - an.y) / an.w / VXY_;
      float tw = __logf(gb.z / an.z) / VWH_;
      float th = __logf(gb.w / an.w) / VWH_;
      acc_reg += sl1f(bp.x - tx) + sl1f(bp.y - ty) + sl1f(bp.z - tw) + sl1f(bp.w - th);
    }
    if (neg) acc_nn += 1;
    __syncthreads();                           // done reading buf before reuse
  }

  // deterministic block reductions -> fixed partial slots (no atomics)
  redf[tid] = acc_pce; __syncthreads();
  for (int s = THR_C/2; s > 0; s >>= 1) { if (tid < s) redf[tid] += redf[tid+s]; __syncthreads(); }
  if (tid == 0) p_posce[b * NBLK_ + blk] = redf[0];
  __syncthreads();
  redf[tid] = acc_reg; __syncthreads();
  for (int s = THR_C/2; s > 0; s >>= 1) { if (tid < s) redf[tid] += redf[tid+s]; __syncthreads(); }
  if (tid == 0) p_reg[b * NBLK_ + blk] = redf[0];
  __syncthreads();
  redi[tid] = acc_np; __syncthreads();
  for (int s = THR_C/2; s > 0; s >>= 1) { if (tid < s) redi[tid] += redi[tid+s]; __syncthreads(); }
  if (tid == 0) p_npos[b * NBLK_ + blk] = redi[0];
  __syncthreads();
  redi[tid] = acc_nn; __syncthreads();
  for (int s = THR_C/2; s > 0; s >>= 1) { if (tid < s) redi[tid] += redi[tid+s]; __syncthreads(); }
  if (tid == 0) p_nneg[b * NBLK_ + blk] = redi[0];
}

// ---------------- kernel 4: exact top-k sums via radix select + per-batch loss ----------------
__global__ __launch_bounds__(256)
void finalize_kernel(const float* __restrict__ ce_sel,
                     const float* __restrict__ p_posce, const float* __restrict__ p_reg,
                     const int* __restrict__ p_npos, const int* __restrict__ p_nneg,
                     float* __restrict__ r_cls, float* __restrict__ r_reg,
                     float* __restrict__ r_np) {
  __shared__ unsigned int hist[256];
  __shared__ float redf[256];
  __shared__ unsigned int sh_prefix, sh_above; __shared__ int sh_k;
  __shared__ float sh_top[2];
  __shared__ int s_np, s_nn; __shared__ float s_pce, s_reg;
  const int b = blockIdx.x, tid = threadIdx.x;

  if (tid == 0) {
    int np = 0, nn = 0; float pc = 0.f, rg = 0.f;
    for (int i = 0; i < NBLK_; ++i) {          // fixed order -> deterministic
      np += p_npos[b*NBLK_+i]; nn += p_nneg[b*NBLK_+i];
      pc += p_posce[b*NBLK_+i]; rg += p_reg[b*NBLK_+i];
    }
    s_np = np; s_nn = nn; s_pce = pc; s_reg = rg;
  }
  __syncthreads();
  const int npos = s_np, nneg = s_nn;
  const float posce = s_pce, regsum = s_reg;
  const float* ce = ce_sel + (long)b * NA_;
  int ks[2] = { min(NPR_ * npos, nneg), min(100, nneg) };

  for (int sel = 0; sel < 2; ++sel) {
    int k = ks[sel];
    float topsum = 0.f;
    if (k > 0) {
      unsigned prefix = 0, above = 0; int kcur = k;
      for (int shift = 24; shift >= 0; shift -= 8) {
        hist[tid] = 0; __syncthreads();
        for (int i = tid; i < NA_; i += 256) {
          unsigned key = __float_as_uint(ce[i]);   // all keys >= 0 -> monotone bits
          int hi = shift + 8;
          bool cand = (hi >= 32) ? true : ((key >> hi) == (prefix >> hi));
          if (cand) atomicAdd(&hist[(key >> shift) & 255], 1u);
        }
        __syncthreads();
        if (tid == 0) {
          unsigned cum = 0; int bin = 0;
          for (int j = 255; j >= 0; --j) {
            if (cum + hist[j] >= (unsigned)kcur) { bin = j; break; }
            cum += hist[j];
          }
          sh_prefix = prefix | ((unsigned)bin << shift);
          sh_k = kcur - (int)cum;
          sh_above = above + cum;
        }
        __syncthreads();
        prefix = sh_prefix; kcur = sh_k; above = sh_above;
        __syncthreads();
      }
      // sum of values strictly above threshold + ties at exact threshold value
      float part = 0.f;
      for (int i = tid; i < NA_; i += 256) {
        unsigned key = __float_as_uint(ce[i]);
        if (key > prefix) part += ce[i];
      }
      redf[tid] = part; __syncthreads();
      for (int s = 128; s > 0; s >>= 1) { if (tid < s) redf[tid] += redf[tid+s]; __syncthreads(); }
      topsum = redf[0] + __uint_as_float(prefix) * (float)(k - (int)above);
      __syncthreads();
    }
    if (tid == 0) sh_top[sel] = topsum;
    __syncthreads();
  }

  if (tid == 0) {
    float top1 = sh_top[0], top2 = sh_top[1];
    int k1 = ks[0], k2 = ks[1];
    float npf = (float)npos;
    float cls_pos  = (k1 > 0) ? (posce + top1) / fmaxf(npf + (float)k1, 1.f)
                              : posce / fmaxf(npf, 1.f);
    float cls_zero = (k2 > 0) ? top2 / fmaxf((float)k2, 1.f) : 0.f;
    r_cls[b] = (npos > 0) ? cls_pos : cls_zero;
    r_reg[b] = (npos > 0) ? regsum / fmaxf(npf * 4.f, 1.f) : 0.f;
    r_np[b]  = npf;
  }
}

// ---------------- kernel 5: final means ----------------
__global__ void mean_kernel(const float* r_cls, const float* r_reg, const float* r_np,
                            float* out) {
  if (threadIdx.x == 0) {
    float a = 0.f, c = 0.f, n = 0.f;
    for (int i = 0; i < B_; ++i) { a += r_cls[i]; c += r_reg[i]; n += r_np[i]; }
    out[0] = a / (float)B_; out[1] = c / (float)B_; out[2] = n / (float)B_;
  }
}

// ---------------- launcher ----------------
extern "C" void kernel_launch(void* const* d_in, const int* in_sizes, int n_in,
                              void* d_out, int out_size, void* d_ws, size_t ws_size,
                              hipStream_t stream) {
  const float* cls_pred  = (const float*)d_in[0];
  const float* box_pred  = (const float*)d_in[1];
  const float* anchors   = (const float*)d_in[2];
  const float* gt_boxes  = (const float*)d_in[3];
  const int*   gt_labels = (const int*)d_in[4];
  float* out = (float*)d_out;

  char* ws = (char*)d_ws;
  size_t off = 0;
  auto alloc = [&](size_t bytes) -> void* {
    void* p = ws + off; off += (bytes + 255) & ~(size_t)255; return p;
  };
  unsigned long long* best = (unsigned long long*)alloc((size_t)B_*NG_*8);
  unsigned char* meta      = (unsigned char*)alloc((size_t)B_*NA_);
  float* ce_sel            = (float*)alloc((size_t)B_*NA_*4);
  float* p_posce           = (float*)alloc((size_t)B_*NBLK_*4);
  float* p_reg             = (float*)alloc((size_t)B_*NBLK_*4);
  int*   p_npos            = (int*)alloc((size_t)B_*NBLK_*4);
  int*   p_nneg            = (int*)alloc((size_t)B_*NBLK_*4);
  float* r_cls             = (float*)alloc((size_t)B_*4);
  float* r_reg             = (float*)alloc((size_t)B_*4);
  float* r_np              = (float*)alloc((size_t)B_*4);

  hipLaunchKernelGGL(init_kernel, dim3(1), dim3(256), 0, stream, best);
  hipLaunchKernelGGL(match_kernel, dim3(NA_/256, B_), dim3(256), 0, stream,
                     anchors, gt_boxes, best, meta);
  hipLaunchKernelGGL(override_kernel, dim3((B_*NG_+255)/256), dim3(256), 0, stream,
                     best, (unsigned int*)meta);
  (void)hipFuncSetAttribute((const void*)ce_kernel,
                            hipFuncAttributeMaxDynamicSharedMemorySize,
                            2 * TILE_F * 4);
  hipLaunchKernelGGL(ce_kernel, dim3(NBLK_, B_), dim3(THR_C), 2*TILE_F*4, stream,
                     cls_pred, box_pred, anchors, gt_boxes, gt_labels, meta,
                     ce_sel, p_posce, p_reg, p_npos, p_nneg);
  hipLaunchKernelGGL(finalize_kernel, dim3(B_), dim3(256), 0, stream,
                     ce_sel, p_posce, p_reg, p_npos, p_nneg, r_cls, r_reg, r_np);
  hipLaunchKernelGGL(mean_kernel, dim3(1), dim3(32), 0, stream, r_cls, r_reg, r_np, out);
}